// BAGLayer_68702296867335
// MI455X (gfx1250) — compile-verified
//
#include <hip/hip_runtime.h>
#include <hip/hip_bf16.h>

typedef float v2f __attribute__((ext_vector_type(2)));
typedef float v8f __attribute__((ext_vector_type(8)));

#define RADIUS2f (10000.0f * 10000.0f)

constexpr int Nn = 8192;   // query points
constexpr int Mn = 8192;   // all points
constexpr int Cn = 6;      // channels
constexpr int CP = 8;      // channels padded to 2 WMMA k-steps of 4
constexpr int Dn = 256;    // hidden dim
constexpr int Kn = 32;     // neighbors / logits
constexpr int WAVES = 8;   // waves (= points) per workgroup

__device__ __forceinline__ void wave_dswait() {
  // intra-wave LDS RAW ordering (wave32 lockstep; no block barrier needed)
  asm volatile("s_wait_dscnt 0" ::: "memory");
}

__global__ __launch_bounds__(WAVES * 32)
void bag_layer_kernel(const float* __restrict__ x,     // (C,N)
                      const float* __restrict__ ap,    // (C,M)
                      const float* __restrict__ w_c1,  // (D,C)
                      const float* __restrict__ b_c1,  // (D)
                      const float* __restrict__ w_e,   // (D,C)
                      const float* __restrict__ b_e,   // (D)
                      const float* __restrict__ w_n,   // (D,C)
                      const float* __restrict__ b_n,   // (D)
                      const float* __restrict__ w_c2,  // (K,D)
                      const float* __restrict__ b_c2,  // (K)
                      float* __restrict__ out)         // (N,D)
{
  // B-operand panels: weights transposed+padded to (CP x D), shared by block
  __shared__ __align__(16) float s_wn[CP * Dn];
  __shared__ __align__(16) float s_we[CP * Dn];
  // per-wave scratch
  __shared__ __align__(16) float s_en  [WAVES][Kn * CP];  // edge+nei, A layout
  __shared__ __align__(16) float s_edge[WAVES][Kn * CP];  // edge,     A layout
  __shared__ float s_xa [WAVES][Dn];
  __shared__ float s_sn [WAVES][Dn];
  __shared__ float s_se [WAVES][Dn];
  __shared__ float s_att[WAVES][Kn];
  __shared__ int   s_idx[WAVES][Kn];

  const int tid  = threadIdx.x;
  const int wv   = tid >> 5;
  const int lane = tid & 31;
  const int n    = blockIdx.x * WAVES + wv;

  // ---- stage padded weight panels (uniform across block) ----
  for (int t = tid; t < CP * Dn; t += WAVES * 32) {
    int c = t >> 8, d = t & (Dn - 1);
    s_wn[t] = (c < Cn) ? w_n[d * Cn + c] : 0.0f;
    s_we[t] = (c < Cn) ? w_e[d * Cn + c] : 0.0f;
  }
  __syncthreads();

  // ---- ball query: first K indices with dist^2 <= r^2 (ballot compaction) ----
  float xl[Cn];
#pragma unroll
  for (int c = 0; c < Cn; ++c) xl[c] = x[c * Nn + n];

  int found = 0;
  for (int base = 0; base < Mn && found < Kn; base += 32) {
    int j = base + lane;
    float dd = 0.0f;
#pragma unroll
    for (int c = 0; c < Cn; ++c) { float dv = xl[c] - ap[c * Mn + j]; dd += dv * dv; }
    bool pass = (dd <= RADIUS2f);
    unsigned mm = (unsigned)__ballot(pass);
    int slot = found + __popc(mm & ((1u << lane) - 1u));
    if (pass && slot < Kn) s_idx[wv][slot] = j;
    found += __popc(mm);
  }
  wave_dswait();
  int nvalid = found < Kn ? found : Kn;
  int fill = (nvalid > 0) ? s_idx[wv][0] : (Mn - 1);
  if (lane >= nvalid) s_idx[wv][lane] = fill;
  wave_dswait();

  // ---- gather neighbors; edge = log(x - nei); build WMMA A panels ----
  int jk = s_idx[wv][lane];  // lane owns neighbor row k = lane
  float erow[Cn];
#pragma unroll
  for (int c = 0; c < Cn; ++c) {
    float nv = ap[c * Mn + jk];
    float e  = __logf(xl[c] - nv);
    s_edge[wv][lane * CP + c] = e;
    s_en  [wv][lane * CP + c] = e + nv;
    erow[c] = e;
  }
  s_edge[wv][lane * CP + 6] = 0.0f; s_edge[wv][lane * CP + 7] = 0.0f;
  s_en  [wv][lane * CP + 6] = 0.0f; s_en  [wv][lane * CP + 7] = 0.0f;
  wave_dswait();

  // x_before = x + sum_k edge  (butterfly over the 32 neighbor rows)
  float xb[Cn];
#pragma unroll
  for (int c = 0; c < Cn; ++c) {
    float s = erow[c];
    for (int off = 16; off > 0; off >>= 1) s += __shfl_xor(s, off, 32);
    xb[c] = xl[c] + s;
  }

  const int halfsel = (lane & 16) ? 2 : 0;  // K-offset for this lane half
  const int col     = lane & 15;            // N (column) index in 16-wide tile
  const int rbase   = (lane & 16) ? 8 : 0;  // M base for C/D accumulator regs

  // ---- pass 1: evf/ef = relu(A @ Wpad + b); keep only column sums ----
  for (int dt = 0; dt < Dn / 16; ++dt) {
    v8f accn0 = {}, accn1 = {}, acce0 = {}, acce1 = {};
#pragma unroll
    for (int ks = 0; ks < 2; ++ks) {
      const int c0 = ks * 4 + halfsel;
      const int d  = dt * 16 + col;
      v2f bn_, be_;
      bn_.x = s_wn[c0 * Dn + d]; bn_.y = s_wn[(c0 + 1) * Dn + d];
      be_.x = s_we[c0 * Dn + d]; be_.y = s_we[(c0 + 1) * Dn + d];
      v2f an0 = *(const v2f*)&s_en  [wv][( 0 + col) * CP + c0];
      v2f an1 = *(const v2f*)&s_en  [wv][(16 + col) * CP + c0];
      v2f ae0 = *(const v2f*)&s_edge[wv][( 0 + col) * CP + c0];
      v2f ae1 = *(const v2f*)&s_edge[wv][(16 + col) * CP + c0];
      accn0 = __builtin_amdgcn_wmma_f32_16x16x4_f32(false, an0, false, bn_, (short)0, accn0, false, false);
      accn1 = __builtin_amdgcn_wmma_f32_16x16x4_f32(false, an1, false, bn_, (short)0, accn1, false, false);
      acce0 = __builtin_amdgcn_wmma_f32_16x16x4_f32(false, ae0, false, be_, (short)0, acce0, false, false);
      acce1 = __builtin_amdgcn_wmma_f32_16x16x4_f32(false, ae1, false, be_, (short)0, acce1, false, false);
    }
    const int d = dt * 16 + col;
    const float biasn = b_n[d], biase = b_e[d];
    float csn = 0.0f, cse = 0.0f;
#pragma unroll
    for (int r = 0; r < 8; ++r) {
      csn += fmaxf(accn0[r] + biasn, 0.0f) + fmaxf(accn1[r] + biasn, 0.0f);
      cse += fmaxf(acce0[r] + biase, 0.0f) + fmaxf(acce1[r] + biase, 0.0f);
    }
    csn += __shfl_xor(csn, 16, 32);  // combine M halves of same column
    cse += __shfl_xor(cse, 16, 32);
    if (lane < 16) { s_sn[wv][d] = csn; s_se[wv][d] = cse; }
  }
  wave_dswait();

  // ---- xa[d] = relu(x_before . w_c1[d] + b_c1[d]) + sum_evf - sum_ef ----
  for (int i = 0; i < Dn / 32; ++i) {
    int d = i * 32 + lane;
    float a = b_c1[d];
#pragma unroll
    for (int c = 0; c < Cn; ++c) a += xb[c] * w_c1[d * Cn + c];
    s_xa[wv][d] = fmaxf(a, 0.0f) + s_sn[wv][d] - s_se[wv][d];
  }
  wave_dswait();

  // ---- logits (lane = k'), relu, softmax across the 32 lanes ----
  float lg = b_c2[lane];
  for (int d = 0; d < Dn; d += 4) {
    lg += s_xa[wv][d + 0] * w_c2[lane * Dn + d + 0]
        + s_xa[wv][d + 1] * w_c2[lane * Dn + d + 1]
        + s_xa[wv][d + 2] * w_c2[lane * Dn + d + 2]
        + s_xa[wv][d + 3] * w_c2[lane * Dn + d + 3];
  }
  lg = fmaxf(lg, 0.0f);
  float mx = lg;
  for (int off = 16; off > 0; off >>= 1) mx = fmaxf(mx, __shfl_xor(mx, off, 32));
  float ex = __expf(lg - mx);
  float es = ex;
  for (int off = 16; off > 0; off >>= 1) es += __shfl_xor(es, off, 32);
  s_att[wv][lane] = ex / es;
  wave_dswait();

  // ---- pass 2: recompute evf tiles; bound[d] = sum_k att[k]*evf[k][d] ----
  for (int dt = 0; dt < Dn / 16; ++dt) {
    v8f accn0 = {}, accn1 = {};
#pragma unroll
    for (int ks = 0; ks < 2; ++ks) {
      const int c0 = ks * 4 + halfsel;
      const int d  = dt * 16 + col;
      v2f bn_;
      bn_.x = s_wn[c0 * Dn + d]; bn_.y = s_wn[(c0 + 1) * Dn + d];
      v2f an0 = *(const v2f*)&s_en[wv][( 0 + col) * CP + c0];
      v2f an1 = *(const v2f*)&s_en[wv][(16 + col) * CP + c0];
      accn0 = __builtin_amdgcn_wmma_f32_16x16x4_f32(false, an0, false, bn_, (short)0, accn0, false, false);
      accn1 = __builtin_amdgcn_wmma_f32_16x16x4_f32(false, an1, false, bn_, (short)0, accn1, false, false);
    }
    const int d = dt * 16 + col;
    const float biasn = b_n[d];
    float bsum = 0.0f;
#pragma unroll
    for (int r = 0; r < 8; ++r) {
      bsum += s_att[wv][ 0 + rbase + r] * fmaxf(accn0[r] + biasn, 0.0f);
      bsum += s_att[wv][16 + rbase + r] * fmaxf(accn1[r] + biasn, 0.0f);
    }
    bsum += __shfl_xor(bsum, 16, 32);  // combine M halves of same column
    if (lane < 16) out[n * Dn + d] = bsum;
  }
}

extern "C" void kernel_launch(void* const* d_in, const int* in_sizes, int n_in,
                              void* d_out, int out_size, void* d_ws, size_t ws_size,
                              hipStream_t stream) {
  (void)in_sizes; (void)n_in; (void)out_size; (void)d_ws; (void)ws_size;
  const float* x    = (const float*)d_in[0];
  const float* ap   = (const float*)d_in[1];
  const float* w_c1 = (const float*)d_in[2];
  const float* b_c1 = (const float*)d_in[3];
  const float* w_e  = (const float*)d_in[4];
  const float* b_e  = (const float*)d_in[5];
  const float* w_n  = (const float*)d_in[6];
  const float* b_n  = (const float*)d_in[7];
  const float* w_c2 = (const float*)d_in[8];
  const float* b_c2 = (const float*)d_in[9];
  float* out = (float*)d_out;

  dim3 grid(Nn / WAVES);   // 1024 workgroups, one point per wave
  dim3 block(WAVES * 32);  // 8 wave32
  bag_layer_kernel<<<grid, block, 0, stream>>>(x, ap, w_c1, b_c1, w_e, b_e,
                                               w_n, b_n, w_c2, b_c2, out);
}